// ResidualDenseBlock_65481071407600
// MI455X (gfx1250) — compile-verified
//
#include <hip/hip_runtime.h>
#include <stdint.h>

// ---------------------------------------------------------------------------
// ResidualDenseBlock: out = scan_{l=0..63}( tanh(out @ W[l]^T + b[l])*m[l] + out ) * sum(gates)
// Memory-streaming + hw-tanh bound. Params (2KB) staged to LDS via CDNA5 async
// global->LDS DMA; hot loop is FMA + v_tanh_f32 per lane-private rows.
// ---------------------------------------------------------------------------

typedef float v4f __attribute__((ext_vector_type(4)));

#if __has_builtin(__builtin_amdgcn_tanhf)
__device__ __forceinline__ float htanh(float x) { return __builtin_amdgcn_tanhf(x); }
#elif __has_builtin(__builtin_amdgcn_tanh_f32)
__device__ __forceinline__ float htanh(float x) { return __builtin_amdgcn_tanh_f32(x); }
#else
__device__ __forceinline__ float htanh(float x) { return tanhf(x); }
#endif

#define LAYERS 64
#define TPB    256
#define IPT    4          // float4 items per thread (= 8 rows/thread)

__global__ __launch_bounds__(TPB)
void rdb_kernel(const float* __restrict__ x,
                const float* __restrict__ W,
                const float* __restrict__ b,
                const float* __restrict__ masks,
                const float* __restrict__ gates,
                float* __restrict__ out,
                int nitems /* number of float4 items */)
{
    // LDS param block: P[l*2+0] = {w00,w01,w10,w11}, P[l*2+1] = {b0,b1,mask,gate}
    __shared__ v4f P[LAYERS * 2];

    const int tid = threadIdx.x;

    // ---- Stage 2KB of params into LDS with async global->LDS DMA ----------
    {
        unsigned ldsBase = (unsigned)(uintptr_t)&P[0];   // low 32 bits = LDS offset
        for (int e = tid; e < LAYERS * 8; e += TPB) {
            const int layer = e >> 3;
            const int slot  = e & 7;
            const float* src;
            if (slot < 4)       src = W     + layer * 4 + slot;        // w00,w01,w10,w11
            else if (slot < 6)  src = b     + layer * 2 + (slot - 4);  // b0,b1
            else if (slot == 6) src = masks + layer;                   // mask
            else                src = gates + layer;                   // gate
            unsigned dst = ldsBase + (unsigned)e * 4u;
            asm volatile("global_load_async_to_lds_b32 %0, %1, off"
                         :: "v"(dst), "v"((unsigned long long)(uintptr_t)src)
                         : "memory");
        }
        asm volatile("s_wait_asynccnt 0" ::: "memory");
        __syncthreads();
    }

    // ---- Load 4 float4 items (8 rows) per thread, fully coalesced ---------
    const v4f* __restrict__ xin = (const v4f*)x;
    v4f*       __restrict__ xo  = (v4f*)out;

    const long base = (long)blockIdx.x * (TPB * IPT) + tid;

    v4f  r[IPT];
    bool valid[IPT];
#pragma unroll
    for (int k = 0; k < IPT; ++k) {
        long idx = base + (long)k * TPB;
        valid[k] = (idx < nitems);
        if (valid[k]) r[k] = __builtin_nontemporal_load(&xin[idx]);
        else          r[k] = (v4f)0.0f;
    }

    // ---- 64-layer recurrence; gate-sum folded into the same loop ----------
    float gs = 0.0f;
#pragma unroll 4
    for (int l = 0; l < LAYERS; ++l) {
        const v4f wv = P[l * 2 + 0];   // ds_load_b128, broadcast (uniform addr)
        const v4f pv = P[l * 2 + 1];
        gs += pv.w;
#pragma unroll
        for (int k = 0; k < IPT; ++k) {
            // rows: (x,y) and (z,w)
            float t0 = htanh(fmaf(wv.y, r[k].y, fmaf(wv.x, r[k].x, pv.x)));
            float t1 = htanh(fmaf(wv.w, r[k].y, fmaf(wv.z, r[k].x, pv.y)));
            float t2 = htanh(fmaf(wv.y, r[k].w, fmaf(wv.x, r[k].z, pv.x)));
            float t3 = htanh(fmaf(wv.w, r[k].w, fmaf(wv.z, r[k].z, pv.y)));
            r[k].x = fmaf(pv.z, t0, r[k].x);
            r[k].y = fmaf(pv.z, t1, r[k].y);
            r[k].z = fmaf(pv.z, t2, r[k].z);
            r[k].w = fmaf(pv.z, t3, r[k].w);
        }
    }

    // ---- Scale by sum(gates) and stream out (nontemporal) -----------------
#pragma unroll
    for (int k = 0; k < IPT; ++k) {
        if (valid[k]) {
            v4f o;
            o.x = r[k].x * gs; o.y = r[k].y * gs;
            o.z = r[k].z * gs; o.w = r[k].w * gs;
            __builtin_nontemporal_store(o, &xo[base + (long)k * TPB]);
        }
    }
}

extern "C" void kernel_launch(void* const* d_in, const int* in_sizes, int n_in,
                              void* d_out, int out_size, void* d_ws, size_t ws_size,
                              hipStream_t stream)
{
    (void)n_in; (void)out_size; (void)d_ws; (void)ws_size;
    const float* x     = (const float*)d_in[0];
    const float* W     = (const float*)d_in[1];
    const float* b     = (const float*)d_in[2];
    const float* masks = (const float*)d_in[3];
    const float* gates = (const float*)d_in[4];
    float*       out   = (float*)d_out;

    const int nfloat = in_sizes[0];        // BATCH * 2  (divisible by 4)
    const int nitems = nfloat / 4;         // float4 count
    const int blocks = (nitems + TPB * IPT - 1) / (TPB * IPT);

    rdb_kernel<<<blocks, TPB, 0, stream>>>(x, W, b, masks, gates, out, nitems);
}